// RelativePositionEncoding_40252433498313
// MI455X (gfx1250) — compile-verified
//
#include <hip/hip_runtime.h>

// CDNA5 WMMA / vector types (ext_vector so "v" asm constraints map to VGPR ranges)
typedef __attribute__((ext_vector_type(2))) float v2f;   // 16x4 (or 4x16) f32 operand
typedef __attribute__((ext_vector_type(8))) float v8f;   // 16x16 f32 accumulator
typedef __attribute__((ext_vector_type(4))) float f4v;   // 128-bit load/store payload

// out[b,s,e] = E[s%8, e] for out shape [32,4096,256] f32.
// Unique content = 8x256 f32 tile (8 KB); output = tile repeated 16384x -> 128 MiB stores.
// Store-bound: ~134 MB / 23.3 TB/s ~= 5.8 us HBM floor; output fits in the 192 MB L2,
// so stores use th:TH_STORE_WB (force dirty in far cache, override wr-rinse) to keep the
// output L2-resident and take the HBM drain off the critical path.
__global__ __launch_bounds__(256) void rpe_bcast_kernel(const float* __restrict__ E,
                                                        float* __restrict__ out,
                                                        int total4) {
    __shared__ float tile[16 * 256];          // 16 KB: two vertical repeats of the 8x256 pattern

    const int tid  = threadIdx.x;
    const int lane = tid & 31;
    const int wave = tid >> 5;                // 8 waves per block

    // --- Stage 1: exact one-hot WMMA gather into LDS ---------------------
    // D[m,n] = sum_k A[m,k] * E_blk[k,n]; A is one-hot f32 so every product is
    // 1.0*x or 0*x and sums are exact under RNE -> bit-identical to a gather.
    const int n     = lane & 15;              // column within 16-wide tile
    const int m     = lane & 15;              // A/D row index held by this lane
    const int kbase = (lane < 16) ? 0 : 2;
    const int mrow  = m & 7;                  // which of the 8 table rows this output row selects

    #pragma unroll
    for (int half = 0; half < 2; ++half) {
        const int n0 = wave * 32 + half * 16; // this wave's 16-column slice of [0,256)

        v2f a1, a2, b1, b2;
        #pragma unroll
        for (int c = 0; c < 2; ++c) {
            const int k = kbase + c;          // 0..3
            a1[c] = (mrow == k)     ? 1.0f : 0.0f;
            a2[c] = (mrow == k + 4) ? 1.0f : 0.0f;
            b1[c] = E[k * 256 + n0 + n];          // table rows 0..3
            b2[c] = E[(k + 4) * 256 + n0 + n];    // table rows 4..7
        }

        v8f d = {};
        d = __builtin_amdgcn_wmma_f32_16x16x4_f32(false, a1, false, b1,
                                                  (short)0, d, false, false);
        d = __builtin_amdgcn_wmma_f32_16x16x4_f32(false, a2, false, b2,
                                                  (short)0, d, false, false);

        // Scatter accumulator to LDS row-major: VGPR c -> row c (lanes 0-15) / c+8 (16-31).
        const int mbase = (lane < 16) ? 0 : 8;
        #pragma unroll
        for (int c = 0; c < 8; ++c) {
            tile[(mbase + c) * 256 + n0 + n] = d[c];
        }
    }

    __syncthreads();

    // --- Stage 2: register-resident float4 streamed to global ------------
    // LDS tile = 1024 float4. Grid stride T is a multiple of 1024 (blocks % 4 == 0),
    // so (p % 1024) is loop-invariant: load the value once, pin it in VGPRs, then the
    // hot loop is pure b128 stores (no ds_load / s_wait_dscnt per iteration).
    const f4v* t4 = (const f4v*)tile;
    f4v*       o4 = (f4v*)out;

    const int g = (int)blockIdx.x * (int)blockDim.x + tid;
    const int T = (int)gridDim.x * (int)blockDim.x;

    f4v val = t4[g & 1023];
    asm volatile("" : "+v"(val));             // opaque: forbid sinking the LDS load into the loop

    #pragma unroll 4
    for (int p = g; p < total4; p += T) {
        f4v* addr = o4 + p;
        // th:TH_STORE_WB -> keep dirty in far cache (192 MB L2 holds the whole 128 MiB output);
        // scope:SCOPE_DEV -> ack from the device-coherent level (global L2).
        asm volatile("global_store_b128 %0, %1, off th:TH_STORE_WB scope:SCOPE_DEV"
                     :: "v"(addr), "v"(val) : "memory");
    }
}

extern "C" void kernel_launch(void* const* d_in, const int* in_sizes, int n_in,
                              void* d_out, int out_size, void* d_ws, size_t ws_size,
                              hipStream_t stream) {
    // setup_inputs order: d_in[0] = x [32,4096,256] f32 (unused by the math),
    //                     d_in[1] = E_relative_position [8,256] f32.
    const float* E   = (const float*)d_in[1];
    float*       out = (float*)d_out;

    const int total4 = out_size / 4;          // 8,388,608 float4 stores

    // 2048 blocks (multiple of 4 -> stride % 1024 == 0) x 256 threads:
    // stride 524,288 float4, exactly 16 b128 stores per thread.
    dim3 grid(2048), block(256);
    rpe_bcast_kernel<<<grid, block, 0, stream>>>(E, out, total4);
}